// SlotGATLayer_90031104459541
// MI455X (gfx1250) — compile-verified
//
#include <hip/hip_runtime.h>
#include <hip/hip_bf16.h>
#include <stdint.h>

// ---------------------------------------------------------------------------
// SlotGAT layer for MI455X (gfx1250, wave32, WMMA + async-LDS copies).
// Problem: C=8, N=1024, T=3, IN_F=64, H=4, OUT_F=64, E=65536.
// Both GEMMs use V_WMMA_F32_16X16X4_F32 (exact fp32 matrix path on CDNA5).
// K4 stages its A tiles with GLOBAL_LOAD_ASYNC_TO_LDS_B128 (ASYNCcnt).
// ---------------------------------------------------------------------------

typedef __attribute__((ext_vector_type(2))) float v2f;
typedef __attribute__((ext_vector_type(8))) float v8f;

#define CC    8
#define NN    1024
#define CN    8192          // C*N
#define TT    3
#define INF   64
#define HH    4
#define FOUT  64
#define HOUT  256           // H*OUT_F
#define KTF   192           // T*OUT_F
#define EE    65536

__device__ __forceinline__ float clipf(float x) {
    return fminf(fmaxf(x, -1.0e9f), 1.0e9f);
}

// ---------------------------------------------------------------------------
// Zero helper (masks live in poisoned workspace).
__global__ void k_zero(float* __restrict__ p, int n) {
    int i = blockIdx.x * blockDim.x + threadIdx.x;
    if (i < n) p[i] = 0.0f;
}

// Scatter edge-endpoint masks: msrc[edge[0,e]]=1, mdst[edge[1,e]]=1.
__global__ void k_edges(const int* __restrict__ eidx,
                        float* __restrict__ msrc, float* __restrict__ mdst) {
    int e = blockIdx.x * blockDim.x + threadIdx.x;
    if (e < EE) {
        msrc[eidx[e]]      = 1.0f;   // all writers store 1.0 -> no atomic needed
        mdst[eidx[EE + e]] = 1.0f;
    }
}

// ---------------------------------------------------------------------------
// K1: feat = einsum('bji,jik->bjk', hs, fc)  (per-t GEMM, M=8192 K=64 N=256),
// stored directly into the torch-permuted layout featP[b][h][t2*64+f2].
// Block = 256 threads (8 waves): 128x16 output tile, wave w -> rows 16w..16w+15.
__global__ __launch_bounds__(256) void k1_gemm_permute(
    const float* __restrict__ node_feats,   // (CN, T, INF)
    const float* __restrict__ fc,           // (T, INF, HOUT)
    float* __restrict__ featP)              // (CN, H, KTF) permuted
{
    __shared__ float As[128][INF + 4];      // stride 68 -> bank-conflict free
    __shared__ float Bs[INF][17];

    const int tid = threadIdx.x;
    const int b0  = blockIdx.x * 128;       // row block (64 blocks)
    const int f0  = blockIdx.y * 16;        // col tile  (16 tiles)
    const int tz  = blockIdx.z;             // t (3)

    // Stage A: 128x64 (float4 per thread x 8)
    {
        int row = tid >> 4;
        int col = (tid & 15) * 4;
        #pragma unroll
        for (int i = 0; i < 8; ++i) {
            const float4 v = *(const float4*)(node_feats +
                (size_t)(b0 + row) * (TT * INF) + tz * INF + col);
            As[row][col + 0] = v.x; As[row][col + 1] = v.y;
            As[row][col + 2] = v.z; As[row][col + 3] = v.w;
            row += 16;
        }
    }
    // Stage B: 64x16 (one float4 per thread)
    {
        int row = tid >> 2;
        int cg  = (tid & 3) * 4;
        const float4 v = *(const float4*)(fc +
            (size_t)tz * (INF * HOUT) + (size_t)row * HOUT + f0 + cg);
        Bs[row][cg + 0] = v.x; Bs[row][cg + 1] = v.y;
        Bs[row][cg + 2] = v.z; Bs[row][cg + 3] = v.w;
    }
    __syncthreads();

    const int wave  = tid >> 5;
    const int lane  = tid & 31;
    const int lhalf = lane >> 4;            // 0|1
    const int l16   = lane & 15;
    const int base  = 2 * lhalf;            // K sub-pair per ISA A/B layout

    v8f acc = {};
    #pragma unroll
    for (int kk = 0; kk < INF; kk += 4) {
        v2f a, b;
        a.x = As[wave * 16 + l16][kk + base];
        a.y = As[wave * 16 + l16][kk + base + 1];
        b.x = Bs[kk + base][l16];
        b.y = Bs[kk + base + 1][l16];
        acc = __builtin_amdgcn_wmma_f32_16x16x4_f32(
                  false, a, false, b, (short)0, acc, false, false);
    }

    // Scatter through the permutation bijection:
    // flat = t*2^18 + n_old*256 + k_old ; dest (n2,t2,h2,f2) from flat.
    #pragma unroll
    for (int r = 0; r < 8; ++r) {
        int gb   = b0 + wave * 16 + (r + 8 * lhalf);   // global row b
        int c    = gb >> 10;
        int nold = gb & 1023;
        int kold = f0 + l16;
        int flat = tz * 262144 + nold * 256 + kold;
        int n2   = flat / 768;
        int rr   = flat - n2 * 768;
        int t2   = rr >> 8;
        int r2   = rr & 255;
        int h2   = r2 >> 6;
        int f2   = r2 & 63;
        featP[(size_t)(((c << 10) | n2) * HH + h2) * KTF + t2 * FOUT + f2] = acc[r];
    }
}

// ---------------------------------------------------------------------------
// K2: a_src[c,n,h] = msrc[b] * sum_k clip(featP[b,h,k]*attn_src[h,k]); same dst.
// One wave per (b,h): 6 strided loads/lane, shfl_xor tree reduce.
__global__ __launch_bounds__(128) void k2_attn(
    const float* __restrict__ featP,
    const float* __restrict__ attn_src, const float* __restrict__ attn_dst,
    const float* __restrict__ msrc, const float* __restrict__ mdst,
    float* __restrict__ aS, float* __restrict__ aD)
{
    int gw   = blockIdx.x * (blockDim.x >> 5) + (threadIdx.x >> 5);
    int lane = threadIdx.x & 31;
    int b    = gw >> 2;
    int h    = gw & 3;

    const float* fp  = featP + (size_t)(b * HH + h) * KTF;
    const float* asv = attn_src + h * KTF;
    const float* adv = attn_dst + h * KTF;

    float ss = 0.0f, sd = 0.0f;
    #pragma unroll
    for (int i = 0; i < 6; ++i) {
        int k = lane + 32 * i;
        float p = fp[k];
        ss += clipf(p * asv[k]);
        sd += clipf(p * adv[k]);
    }
    #pragma unroll
    for (int off = 16; off > 0; off >>= 1) {
        ss += __shfl_xor(ss, off, 32);
        sd += __shfl_xor(sd, off, 32);
    }
    if (lane == 0) {
        aS[b * HH + h] = ss * msrc[b];
        aD[b * HH + h] = sd * mdst[b];
    }
}

// ---------------------------------------------------------------------------
// K3: a[c,v,u,h] = softmax_c(clip(lrelu(a_src[c,v,h]+a_dst[c,u,h]))). Writes
// the (c,n,n,h) output tensor; 8-wide softmax kept in registers.
__global__ __launch_bounds__(256) void k3_softmax(
    const float* __restrict__ aS, const float* __restrict__ aD,
    float* __restrict__ aOut)
{
    int id = blockIdx.x * 256 + threadIdx.x;      // (v,u,h), h fastest
    int h  = id & 3;
    int u  = (id >> 2) & 1023;
    int v  = id >> 12;

    float s[CC];
    float mx = -3.0e38f;
    #pragma unroll
    for (int c = 0; c < CC; ++c) {
        float x = aS[(c << 12) + (v << 2) + h] + aD[(c << 12) + (u << 2) + h];
        x = (x >= 0.0f) ? x : 0.2f * x;           // leaky relu
        x = clipf(x);
        s[c] = x;
        mx = fmaxf(mx, x);
    }
    float sum = 0.0f;
    #pragma unroll
    for (int c = 0; c < CC; ++c) { s[c] = expf(s[c] - mx); sum += s[c]; }
    float inv = 1.0f / sum;
    #pragma unroll
    for (int c = 0; c < CC; ++c) {
        aOut[((size_t)((c << 10) + v) * 1024 + u) * 4 + h] = s[c] * inv;
    }
}

// ---------------------------------------------------------------------------
// K4: m[c,v,h,f] = sum_u a[c,v,u,h] * featP[c*N+u,h,f] * mdst[c*N+u];
// outs[c,v,t,f2] = sum_h relu(m). Block = 256 thr (8 waves) computes a
// 128x16 tile for fixed (c, f-tile) and ALL FOUR heads at once: the four h
// values of one (v,u) are contiguous in `a`, so the A tile is staged with
// coalesced GLOBAL_LOAD_ASYNC_TO_LDS_B128 (one 16B packet = 4 heads), and the
// relu+sum_h epilogue is fused in-registers (no atomics -> deterministic).
__global__ __launch_bounds__(256) void k4_gemm_out(
    const float* __restrict__ aIn,      // (C,N,N,H)
    const float* __restrict__ featP,    // (CN,H,KTF)
    const float* __restrict__ mdst,     // (CN)
    float* __restrict__ outs)           // (C,N,T,FOUT)
{
    // As[row][u*4 + h]: row stride 132 floats (33x16B) -> 16B-aligned slots and
    // conflict-free column reads (132 mod 64 = 4 banks/row).
    __shared__ __attribute__((aligned(16))) float As[128][132];
    __shared__ float Bs[HH][32][17];

    const int tid = threadIdx.x;
    const int f0  = blockIdx.x * 16;    // 12 f-tiles
    const int v0  = blockIdx.y * 128;   // 8 v-blocks
    const int c   = blockIdx.z;         // 8 conformers

    const int wave  = tid >> 5;
    const int lane  = tid & 31;
    const int lhalf = lane >> 4;
    const int l16   = lane & 15;
    const int base  = 2 * lhalf;

    // Uniform 64-bit base for the async copies (SGPR pair); per-lane 32-bit
    // byte offsets (c-slice is 16 MB, fits signed 32-bit).
    const float* aC = aIn + (size_t)c * NN * NN * HH;

    // A staging geometry: thread -> row = tid/2 (0..127), half of 32 u-groups.
    const int arow = tid >> 1;
    const int aug0 = (tid & 1) * 16;

    v8f acc[HH];
    #pragma unroll
    for (int h = 0; h < HH; ++h) acc[h] = (v8f){};

    for (int u0 = 0; u0 < NN; u0 += 32) {
        __syncthreads();   // previous-iteration LDS readers done

        // Async-stage A: 128 rows x 32 u, 16B per (v,u) = all 4 heads.
        #pragma unroll
        for (int j = 0; j < 16; ++j) {
            int ug = aug0 + j;
            uint32_t voff  = (uint32_t)(((v0 + arow) * 1024u + (u0 + ug)) * 16u);
            uint32_t laddr = (uint32_t)(uintptr_t)&As[arow][ug * 4];
            asm volatile("global_load_async_to_lds_b128 %0, %1, %2"
                         :: "v"(laddr), "v"(voff), "s"(aC) : "memory");
        }

        // Stage B (32u x 16f per head) with mdst row-scale folded in.
        #pragma unroll
        for (int i = 0; i < 8; ++i) {
            int e  = tid + 256 * i;
            int h  = e >> 9;
            int uu = (e >> 4) & 31;
            int f  = e & 15;
            int gu = (c << 10) + u0 + uu;
            Bs[h][uu][f] = featP[(size_t)(gu * HH + h) * KTF + f0 + f] * mdst[gu];
        }

        asm volatile("s_wait_asynccnt 0x0" ::: "memory");
        __syncthreads();

        #pragma unroll
        for (int h = 0; h < HH; ++h) {
            #pragma unroll
            for (int kk = 0; kk < 32; kk += 4) {
                v2f a, b;
                a.x = As[wave * 16 + l16][(kk + base)     * 4 + h];
                a.y = As[wave * 16 + l16][(kk + base + 1) * 4 + h];
                b.x = Bs[h][kk + base][l16];
                b.y = Bs[h][kk + base + 1][l16];
                acc[h] = __builtin_amdgcn_wmma_f32_16x16x4_f32(
                             false, a, false, b, (short)0, acc[h], false, false);
            }
        }
    }

    // Epilogue: outs[c,v,t,f2] = sum_h relu(m_h); f-tile of 16 lies inside one t.
    v8f osum = {};
    #pragma unroll
    for (int h = 0; h < HH; ++h)
        #pragma unroll
        for (int r = 0; r < 8; ++r) osum[r] += fmaxf(acc[h][r], 0.0f);

    const int f  = f0 + l16;
    const int t  = f >> 6;
    const int f2 = f & 63;
    #pragma unroll
    for (int r = 0; r < 8; ++r) {
        int v = v0 + wave * 16 + (r + 8 * lhalf);
        outs[(size_t)(((c << 10) + v) * TT + t) * FOUT + f2] = osum[r];
    }
}

// ---------------------------------------------------------------------------
extern "C" void kernel_launch(void* const* d_in, const int* in_sizes, int n_in,
                              void* d_out, int out_size, void* d_ws, size_t ws_size,
                              hipStream_t stream) {
    const int*   edge       = (const int*)d_in[0];     // (2, E)
    const float* node_feats = (const float*)d_in[1];   // (C,N,T,IN_F)
    const float* fc         = (const float*)d_in[2];   // (T,IN_F,H*OUT_F)
    const float* attn_src   = (const float*)d_in[3];   // (1,H,T*OUT_F)
    const float* attn_dst   = (const float*)d_in[4];   // (1,H,T*OUT_F)

    float* outs = (float*)d_out;                       // C*N*T*FOUT = 1572864
    float* aOut = (float*)d_out + (size_t)CC * NN * TT * FOUT;

    // Workspace layout (floats): featP | msrc | mdst | aS | aD  (~25.5 MB)
    float* ws    = (float*)d_ws;
    float* featP = ws;                                   // CN*H*KTF = 6291456
    float* msrc  = featP + (size_t)CN * HH * KTF;        // 8192
    float* mdst  = msrc + CN;                            // 8192
    float* aS    = mdst + CN;                            // 32768
    float* aD    = aS + CN * HH;                         // 32768

    k_zero<<<(2 * CN + 255) / 256, 256, 0, stream>>>(msrc, 2 * CN);
    k_edges<<<(EE + 255) / 256, 256, 0, stream>>>(edge, msrc, mdst);

    dim3 g1(CN / 128, HOUT / 16, TT);                   // 64 x 16 x 3
    k1_gemm_permute<<<g1, 256, 0, stream>>>(node_feats, fc, featP);

    k2_attn<<<(CN * HH) / 4, 128, 0, stream>>>(featP, attn_src, attn_dst,
                                               msrc, mdst, aS, aD);

    k3_softmax<<<(NN * NN * HH) / 256, 256, 0, stream>>>(aS, aD, aOut);

    dim3 g4(KTF / 16, NN / 128, CC);                    // 12 x 8 x 8
    k4_gemm_out<<<g4, 256, 0, stream>>>(aOut, featP, mdst, outs);
}